// GraphProjection_65635690217825
// MI455X (gfx1250) — compile-verified
//
#include <hip/hip_runtime.h>
#include <math.h>

typedef __attribute__((ext_vector_type(2))) float v2f;
typedef __attribute__((ext_vector_type(8))) float v8f;
typedef int i32x4 __attribute__((vector_size(16)));   // matches builtin param type

#define BATCH 8
#define CCH 512
#define HW 64
#define BNOD 8
#define GBLK 16
#define NPIX 256
#define NODE_NUM 128
#define XSTR 260          // LDS row stride per channel (floats); 260 % 64 == 4 -> bank friendly
#define CH_A 64           // channels per LDS chunk, logits pass
#define CH_C 128          // channels per LDS chunk, aggregation pass
#define EPSV 1e-9f

// ---- LDS layout (float offsets) ----
// Two ping-pong x buffers sized for the 128-channel stage-C chunks; stage A
// uses only the first CH_A*XSTR floats of each. The wu table overlays the
// upper (stage-A-unused) half of buffer 1 and is dead before stage C.
#define OFF_XB0    0
#define OFF_XB1    (CH_C * XSTR)                 // 33280
#define OFF_WU     (OFF_XB1 + CH_A * XSTR)       // 49920: float2[512][16]
#define OFF_SOFT   (2 * CH_C * XSTR)             // 66560: float[256][16]
#define OFF_PFULL  (OFF_SOFT + NPIX * 16)        // float[8][512]
#define OFF_RED    (OFF_PFULL + BNOD * CCH)      // float[256]
#define OFF_Q      (OFF_RED + 256)
#define OFF_Z      (OFF_Q + 8)
#define OFF_N2     (OFF_Z + 8)
#define SMEM_FLOATS (OFF_N2 + 8)                 // 75032 floats = 300128 B < 320 KB

#if __has_builtin(__builtin_amdgcn_global_load_async_to_lds_b128) && \
    __has_builtin(__builtin_amdgcn_s_wait_asynccnt)
#define HAVE_ASYNC 1
#else
#define HAVE_ASYNC 0
#endif

__device__ __forceinline__ void async_copy16(const float* g, float* l) {
#if HAVE_ASYNC
  __builtin_amdgcn_global_load_async_to_lds_b128(
      (__attribute__((address_space(1))) i32x4*)(void*)g,
      (__attribute__((address_space(3))) i32x4*)(void*)l, 0, 0);
#else
  *(float4*)l = *(const float4*)g;
#endif
}

__device__ __forceinline__ void wait_async0() {
#if HAVE_ASYNC
  __builtin_amdgcn_s_wait_asynccnt(0);
#endif
}

// Issue one chunk of x (nch channels starting at cbase) into LDS buffer xb.
__device__ __forceinline__ void issue_chunk(const float* __restrict__ x, float* xb,
                                            int b, int by, int bx, int cbase,
                                            int nch, int tid) {
  const int nslots = nch * 64;                 // float4 slots
  for (int s = tid; s < nslots; s += 256) {
    int c = s >> 6, r = (s >> 2) & 15, jq = s & 3;
    const float* g = x + (((size_t)b * CCH + cbase + c) * HW + by * 16 + r) * HW +
                     bx * 16 + jq * 4;
    async_copy16(g, xb + c * XSTR + r * 16 + jq * 4);
  }
}

__global__ __launch_bounds__(256) void graphproj_kernel(
    const float* __restrict__ x, const float* __restrict__ anchor,
    const float* __restrict__ sigma, float* __restrict__ out_nodes,
    float* __restrict__ out_soft)
{
  extern __shared__ float smem[];
  float*  xb0   = smem + OFF_XB0;
  float*  xb1   = smem + OFF_XB1;
  float2* wubuf = (float2*)(smem + OFF_WU);
  float*  soft  = smem + OFF_SOFT;
  float*  Pfull = smem + OFF_PFULL;
  float*  red   = smem + OFF_RED;
  float*  qbuf  = smem + OFF_Q;
  float*  zbuf  = smem + OFF_Z;
  float*  n2buf = smem + OFF_N2;

  const int tid  = threadIdx.x;
  const int wave = tid >> 5;
  const int lane = tid & 31;
  const int half = lane >> 4;   // 0: lanes 0-15 (K0/K1), 1: lanes 16-31 (K2/K3)
  const int li   = lane & 15;

  const int bg = blockIdx.x;
  const int b  = bg >> 4;
  const int g  = bg & 15;
  const int by = g >> 2, bx = g & 3;

  // ---------- preprocessing: wu = (1/s^2, -2a/s^2) table [c][node], q[k] ----------
  {
    const int k   = tid >> 5;       // node 0..7, 32 threads each
    const int l32 = tid & 31;
    const int gk  = g * BNOD + k;
    float qp = 0.f;
    for (int i = 0; i < 16; ++i) {
      int   c = l32 + 32 * i;
      float a = anchor[gk * CCH + c];
      float s = sigma[gk * CCH + c];
      float w = 1.f / (s * s);
      wubuf[c * 16 + k] = make_float2(w, -2.f * a * w);
      qp += a * a * w;
    }
    red[tid] = qp;
    // zero pad nodes 8..15 so WMMA B-operand reads are branch-free
    for (int e = tid; e < CCH * 8; e += 256) {
      int c = e >> 3, kk = 8 + (e & 7);
      wubuf[c * 16 + kk] = make_float2(0.f, 0.f);
    }
  }
  __syncthreads();
  if (tid < 8) {
    float s = 0.f;
    for (int j = 0; j < 32; ++j) s += red[tid * 32 + j];
    qbuf[tid] = s;
    zbuf[tid] = 0.f;
    n2buf[tid] = 0.f;
  }

  // ---------- stage A: logits GEMM  L[pix,node] = sum_c (x^2*w - 2*x*u) ----------
  // A (16 pix x 4) augmented K = (x^2[c], x[c], x^2[c+1], x[c+1]); B = (w, -2u).
  // Double-buffered async pipeline: prefetch chunk c+1 while WMMAs run on c.
  v8f acc[2] = {{0,0,0,0,0,0,0,0},{0,0,0,0,0,0,0,0}};
  issue_chunk(x, xb0, b, by, bx, 0, CH_A, tid);
  wait_async0();
  __syncthreads();
  for (int chunk = 0; chunk < CCH / CH_A; ++chunk) {
    float* cur = (chunk & 1) ? xb1 : xb0;
    if (chunk + 1 < CCH / CH_A)
      issue_chunk(x, (chunk & 1) ? xb0 : xb1, b, by, bx, (chunk + 1) * CH_A, CH_A, tid);
    for (int t = 0; t < 2; ++t) {
      const int mt   = wave * 2 + t;       // pixel tile 0..15
      const int pixi = mt * 16 + li;
      #pragma unroll 4
      for (int cc = 0; cc < CH_A; cc += 2) {
        float xv = cur[(cc + half) * XSTR + pixi];
        v2f av; av.x = xv * xv; av.y = xv;
        float2 wv = wubuf[(chunk * CH_A + cc + half) * 16 + li];
        v2f bv; bv.x = wv.x; bv.y = wv.y;
        acc[t] = __builtin_amdgcn_wmma_f32_16x16x4_f32(
            false, av, false, bv, (short)0, acc[t], false, false);
      }
    }
    wait_async0();          // drain prefetch (overlapped with the WMMA loop)
    __syncthreads();        // all waves done reading `cur` + chunk+1 visible
  }
  // write logits = -0.5*(acc + q) into soft[][]
  for (int t = 0; t < 2; ++t) {
    const int mt = wave * 2 + t;
    if (li < BNOD) {
      float qv = qbuf[li];
      #pragma unroll
      for (int v = 0; v < 8; ++v) {
        int pix = mt * 16 + v + half * 8;   // D layout: M = v + 8*half
        soft[pix * 16 + li] = -0.5f * (acc[t][v] + qv);
      }
    }
  }
  __syncthreads();

  // ---------- stage B: softmax over 8 nodes per pixel; emit soft_assign; Z[k] ----------
  {
    const int pix = tid;
    float lg[8];
    float mx = -3.0e38f;
    #pragma unroll
    for (int k = 0; k < 8; ++k) { lg[k] = soft[pix * 16 + k]; mx = fmaxf(mx, lg[k]); }
    float sum = 0.f;
    #pragma unroll
    for (int k = 0; k < 8; ++k) { lg[k] = expf(lg[k] - mx); sum += lg[k]; }
    float inv = 1.f / sum;
    #pragma unroll
    for (int k = 0; k < 8; ++k) {
      float p = lg[k] * inv;
      soft[pix * 16 + k] = p;
      out_soft[((size_t)b * NODE_NUM + g * BNOD + k) * NPIX + pix] = p;  // coalesced per k
    }
    #pragma unroll
    for (int k = 8; k < 16; ++k) soft[pix * 16 + k] = 0.f;  // pad nodes
  }
  __syncthreads();
  if (tid < 128) {                       // Z[k] = sum_pix soft
    int k = tid & 7, seg = tid >> 3;
    float s = 0.f;
    for (int i = 0; i < 16; ++i) s += soft[(seg * 16 + i) * 16 + k];
    red[tid] = s;
  }
  __syncthreads();
  if (tid < 8) {
    float s = 0.f;
    for (int j = 0; j < 16; ++j) s += red[j * 8 + tid];
    zbuf[tid] = s;
  }
  __syncthreads();

  // ---------- stage C: P[k,c] = sum_pix soft[pix,k]*x[pix,c]  (soft^T x GEMM) ----------
  // Same async double-buffer pipeline; pass 2 over x is L2-resident (64 MiB < 192 MB).
  issue_chunk(x, xb0, b, by, bx, 0, CH_C, tid);
  wait_async0();
  __syncthreads();
  for (int chunk = 0; chunk < CCH / CH_C; ++chunk) {
    float* cur = (chunk & 1) ? xb1 : xb0;
    if (chunk + 1 < CCH / CH_C)
      issue_chunk(x, (chunk & 1) ? xb0 : xb1, b, by, bx, (chunk + 1) * CH_C, CH_C, tid);
    v8f pacc = {0,0,0,0,0,0,0,0};
    const int cloc = wave * 16 + li;     // this wave owns a 16-channel column
    #pragma unroll 4
    for (int p0 = 0; p0 < NPIX; p0 += 4) {
      v2f av;                                             // A = soft^T (node rows)
      av.x = soft[(p0 + 2 * half)     * 16 + li];
      av.y = soft[(p0 + 2 * half + 1) * 16 + li];
      float2 xv = *(const float2*)(cur + cloc * XSTR + p0 + 2 * half);  // B = X, b64
      v2f bv; bv.x = xv.x; bv.y = xv.y;
      pacc = __builtin_amdgcn_wmma_f32_16x16x4_f32(
          false, av, false, bv, (short)0, pacc, false, false);
    }
    if (!half) {                         // D rows 0..7 (lanes 0-15) are the 8 real nodes
      const int c0 = chunk * CH_C + wave * 16;
      #pragma unroll
      for (int v = 0; v < 8; ++v) Pfull[v * CCH + c0 + li] = pacc[v];
    }
    wait_async0();
    __syncthreads();
  }

  // ---------- finalize: mean residual, L2 norm, write nodes ----------
  float* npart = soft;                   // soft is dead; reuse 2048 floats
  for (int k = 0; k < 8; ++k) {
    const int gk = g * BNOD + k;
    const float Zk = zbuf[k];
    const float denom = Zk + EPSV;
    float acc2 = 0.f;
    for (int c = tid; c < CCH; c += 256) {
      float a = anchor[gk * CCH + c];
      float s = sigma[gk * CCH + c];
      float m = (Pfull[k * CCH + c] - a * Zk) / (s * denom);
      Pfull[k * CCH + c] = m;
      acc2 += m * m;
    }
    npart[k * 256 + tid] = acc2;
  }
  __syncthreads();
  if (tid < 8) {
    float s = 0.f;
    for (int j = 0; j < 256; ++j) s += npart[tid * 256 + j];
    n2buf[tid] = s;
  }
  __syncthreads();
  for (int k = 0; k < 8; ++k) {
    float rn = 1.f / fmaxf(sqrtf(n2buf[k]), 1e-12f);
    for (int c = tid; c < CCH; c += 256)
      out_nodes[((size_t)b * CCH + c) * NODE_NUM + g * BNOD + k] = Pfull[k * CCH + c] * rn;
  }
}

extern "C" void kernel_launch(void* const* d_in, const int* in_sizes, int n_in,
                              void* d_out, int out_size, void* d_ws, size_t ws_size,
                              hipStream_t stream) {
  const float* x      = (const float*)d_in[0];
  const float* anchor = (const float*)d_in[1];
  const float* sigma  = (const float*)d_in[2];
  float* out_nodes = (float*)d_out;                                  // (B, C, 128)
  float* out_soft  = out_nodes + (size_t)BATCH * CCH * NODE_NUM;     // (B, 128, 256)

  const size_t smem = SMEM_FLOATS * sizeof(float);
  (void)hipFuncSetAttribute(reinterpret_cast<const void*>(&graphproj_kernel),
                            hipFuncAttributeMaxDynamicSharedMemorySize, (int)smem);
  graphproj_kernel<<<BATCH * GBLK, 256, smem, stream>>>(x, anchor, sigma, out_nodes, out_soft);
}